// AttributeAttn_34316788695352
// MI455X (gfx1250) — compile-verified
//
#include <hip/hip_runtime.h>
#include <math.h>

// Problem constants (from reference): hidden (N,B,H), encoder (C,B,K), W (H, H+K)
static constexpr int kN = 1024;
static constexpr int kB = 32;
static constexpr int kH = 1024;
static constexpr int kC = 64;
static constexpr int kK = 512;
static constexpr int kHK = kH + kK;      // 1536

// Workspace layout (in floats):
//   [0 .. 1536)            u = W^T v
//   [1536]                 bias = b . v
//   [2048 .. 2048+32768)   hv[n*32+b]
//   [34816 .. 34816+2048)  ev[c*32+b]
static constexpr int WS_U    = 0;
static constexpr int WS_BIAS = 1536;
static constexpr int WS_HV   = 2048;
static constexpr int WS_EV   = 2048 + kN * kB;   // 34816

typedef float v2f __attribute__((ext_vector_type(2)));
typedef float v8f __attribute__((ext_vector_type(8)));

// ---------------------------------------------------------------------------
// Kernel 1: u[j] = sum_h W[h][j] * v[h]  (j in [0,1536)), bias = b.v
// W is row-major (H x 1536): consecutive j are contiguous -> coalesced.
// Blocks 0..5: 256 columns each. Block 6: bias reduction.
// ---------------------------------------------------------------------------
__global__ void __launch_bounds__(256)
proj_u_kernel(const float* __restrict__ W, const float* __restrict__ bvec,
              const float* __restrict__ v, float* __restrict__ ws) {
  const int tid = threadIdx.x;
  if (blockIdx.x < 6) {
    const int j = blockIdx.x * 256 + tid;
    float acc = 0.0f;
#pragma unroll 8
    for (int h = 0; h < kH; ++h) {
      acc = fmaf(W[(size_t)h * kHK + j], v[h], acc);   // v[h] is wave-uniform (s_load)
    }
    ws[WS_U + j] = acc;
  } else {
    __shared__ float red[256];
    float acc = 0.0f;
    for (int h = tid; h < kH; h += 256) acc = fmaf(bvec[h], v[h], acc);
    red[tid] = acc;
    __syncthreads();
    for (int s = 128; s > 0; s >>= 1) {
      if (tid < s) red[tid] += red[tid + s];
      __syncthreads();
    }
    if (tid == 0) ws[WS_BIAS] = red[0];
  }
}

// ---------------------------------------------------------------------------
// GEMV via V_WMMA_F32_16X16X4_F32, one wave per 16 rows, no divergence
// anywhere (EXEC is all-1s by construction, as the WMMA ISA requires).
//
// A fragment (32-bit 16x4, ISA 7.12.2): lane m<16 holds row m cols {k,k+1} in
// VGPR0/1; lanes 16..31 hold cols {k+2,k+3}. One float2 load at
// row*ld + k + 2*(lane>>4) fills it. B fragment: the u-chunk broadcast across
// all 16 N columns uses the identical slot mapping, read from LDS.
// D[m][0] accumulates sum_k A[m][k]*u[k]; column 0 lives in lane 0
// (rows 0..7, VGPR 0..7) and lane 16 (rows 8..15).
//
// Two independent accumulator chains (even/odd k-chunks) double the WMMA
// latency hiding; summed at the end. Bandwidth-bound: 256 B of HBM per WMMA.
// ---------------------------------------------------------------------------
template <int LEN, int UBASE>
__device__ __forceinline__ void gemv16(const float* __restrict__ src,
                                       const float* __restrict__ s_u,
                                       float* __restrict__ dst, int tile) {
  const int lane = threadIdx.x & 31;
  const int m    = lane & 15;
  const int koff = (lane >> 4) << 1;                          // 0 or 2
  const int r0   = tile << 4;
  const float* __restrict__ rowp = src + (size_t)(r0 + m) * LEN + koff;

  v8f acc0 = {};
  v8f acc1 = {};
  for (int k = 0; k < LEN; k += 8) {
    if ((k & 63) == 0 && (k + 256) < LEN)
      __builtin_prefetch(rowp + k + 256, 0, 1);               // global_prefetch_b8
    const v2f a0 = *(const v2f*)(rowp + k);
    const v2f b0 = *(const v2f*)(&s_u[UBASE + k + koff]);     // ds_load_b64
    const v2f a1 = *(const v2f*)(rowp + k + 4);
    const v2f b1 = *(const v2f*)(&s_u[UBASE + k + 4 + koff]);
    acc0 = __builtin_amdgcn_wmma_f32_16x16x4_f32(
        false, a0, false, b0, (short)0, acc0, false, false);
    acc1 = __builtin_amdgcn_wmma_f32_16x16x4_f32(
        false, a1, false, b1, (short)0, acc1, false, false);
  }
  const v8f acc = acc0 + acc1;

  if (lane == 0) {
#pragma unroll
    for (int i = 0; i < 8; ++i) dst[r0 + i] = acc[i];
  }
  if (lane == 16) {
#pragma unroll
    for (int i = 0; i < 8; ++i) dst[r0 + 8 + i] = acc[i];
  }
}

// hv[n*32+b] = hidden[n,b,:] . u[0:1024]   (2048 tiles, 256 blocks x 8 waves)
__global__ void __launch_bounds__(256)
gemv_hidden_kernel(const float* __restrict__ hidden, float* __restrict__ ws) {
  __shared__ __align__(16) float s_u[kH];
  for (int i = threadIdx.x; i < kH; i += 256) s_u[i] = ws[WS_U + i];
  __syncthreads();
  const int tile = (blockIdx.x << 3) + (threadIdx.x >> 5);
  gemv16<kH, 0>(hidden, s_u, ws + WS_HV, tile);
}

// ev[c*32+b] = enc[c,b,:] . u[1024:1536]   (128 tiles, 16 blocks x 8 waves)
__global__ void __launch_bounds__(256)
gemv_enc_kernel(const float* __restrict__ enc, float* __restrict__ ws) {
  __shared__ __align__(16) float s_u[kK];
  for (int i = threadIdx.x; i < kK; i += 256) s_u[i] = ws[WS_U + kH + i];
  __syncthreads();
  const int tile = (blockIdx.x << 3) + (threadIdx.x >> 5);
  gemv16<kK, 0>(enc, s_u, ws + WS_EV, tile);
}

// ---------------------------------------------------------------------------
// Kernel 3: out[b][n][c] = softmax_c( tanh(hv[n,b] + ev[c,b] + bias) )
// One wave per (b,n); each lane covers c=lane and c=lane+32.
// Wave-level max/sum reductions via __shfl_xor (32 lanes, wave32).
// Output stores fully coalesced (64 contiguous floats per wave).
// ---------------------------------------------------------------------------
__global__ void __launch_bounds__(256)
attn_softmax_kernel(const float* __restrict__ ws, float* __restrict__ out) {
  const int wave = (blockIdx.x << 3) + (threadIdx.x >> 5);
  const int lane = threadIdx.x & 31;
  const int n = wave >> 5;          // [0,1024)
  const int b = wave & 31;          // [0,32)

  const float bias = ws[WS_BIAS];
  const float hv   = ws[WS_HV + n * kB + b];
  const float* __restrict__ ev = ws + WS_EV;

  const float t0 = tanhf(hv + ev[lane * kB + b] + bias);
  const float t1 = tanhf(hv + ev[(lane + 32) * kB + b] + bias);

  float mx = fmaxf(t0, t1);
#pragma unroll
  for (int s = 16; s > 0; s >>= 1) mx = fmaxf(mx, __shfl_xor(mx, s, 32));

  const float e0 = __expf(t0 - mx);
  const float e1 = __expf(t1 - mx);
  float sum = e0 + e1;
#pragma unroll
  for (int s = 16; s > 0; s >>= 1) sum += __shfl_xor(sum, s, 32);

  const float inv = 1.0f / sum;
  const size_t base = ((size_t)b * kN + n) * kC;
  out[base + lane]      = e0 * inv;
  out[base + 32 + lane] = e1 * inv;
}

// ---------------------------------------------------------------------------
extern "C" void kernel_launch(void* const* d_in, const int* in_sizes, int n_in,
                              void* d_out, int out_size, void* d_ws, size_t ws_size,
                              hipStream_t stream) {
  const float* hidden = (const float*)d_in[0];   // (1024, 32, 1024)
  const float* enc    = (const float*)d_in[1];   // (64, 32, 512)
  const float* W      = (const float*)d_in[2];   // (1024, 1536)
  const float* bvec   = (const float*)d_in[3];   // (1024,)
  const float* v      = (const float*)d_in[4];   // (1024,)
  float*       out    = (float*)d_out;           // (32, 1024, 64)
  float*       ws     = (float*)d_ws;

  // K1: u and bias (7 blocks: 6 x 256 columns + 1 reduction block)
  proj_u_kernel<<<7, 256, 0, stream>>>(W, bvec, v, ws);

  // K2a: 2048 hidden tiles = 256 blocks x 8 waves (streams 128 MB)
  gemv_hidden_kernel<<<256, 256, 0, stream>>>(hidden, ws);

  // K2b: 128 encoder tiles = 16 blocks x 8 waves (streams 4 MB)
  gemv_enc_kernel<<<16, 256, 0, stream>>>(enc, ws);

  // K3: 32768 (b,n) pairs, 1 wave each -> 4096 blocks x 8 waves (writes 8 MB)
  attn_softmax_kernel<<<4096, 256, 0, stream>>>(ws, out);
}